// RGATLayer_66924180406879
// MI455X (gfx1250) — compile-verified
//
#include <hip/hip_runtime.h>
#include <math.h>

// ---------------- problem constants (from reference) ----------------
constexpr int Bc  = 2;
constexpr int Nc  = 20000;
constexpr int Rc  = 8;
constexpr int Hc  = 4;
constexpr int Cc  = 32;
constexpr int Ac  = 32;
constexpr int INc = 128;
constexpr int NBc = 4;
constexpr int Fc  = Rc * Hc * Cc;          // 1024  (R*H*C)
constexpr int Gc  = Rc * Hc * 2 * Ac;      // 2048  (R*H*2A)
constexpr int Mrows = Bc * Nc;             // 40000
constexpr int Sc  = Bc * Nc * Rc;          // 320000 softmax segments

typedef float v2f __attribute__((ext_vector_type(2)));
typedef float v8f __attribute__((ext_vector_type(8)));

// ---------------- helpers ----------------
__device__ __forceinline__ unsigned flipf(float f) {
    unsigned u = __float_as_uint(f);
    return (u & 0x80000000u) ? ~u : (u | 0x80000000u);
}
__device__ __forceinline__ float unflipf(unsigned u) {
    unsigned v = (u & 0x80000000u) ? (u & 0x7FFFFFFFu) : ~u;
    return __uint_as_float(v);
}

// ---------------- init: zero out/den, set max to flip-space -inf (=0) ----------------
__global__ void init_bufs(float* __restrict__ out, float* __restrict__ den,
                          unsigned* __restrict__ mmax, int nOut, int nSH) {
    int i = blockIdx.x * blockDim.x + threadIdx.x;
    if (i < nOut) out[i] = 0.0f;
    if (i < nSH) { den[i] = 0.0f; mmax[i] = 0u; }
}

// ---------------- W1[p,i] = sum_k proj_coeff[rh,k]*proj_basis[k,c,i] ----------------
__global__ void build_w1(const float* __restrict__ coeff, const float* __restrict__ basis,
                         float* __restrict__ W1) {
    int t = blockIdx.x * blockDim.x + threadIdx.x;  // [0, Fc*INc)
    if (t >= Fc * INc) return;
    int p = t / INc, i = t % INc;
    int rh = p / Cc, c = p % Cc;
    float s = 0.f;
#pragma unroll
    for (int k = 0; k < NBc; ++k)
        s += coeff[rh * NBc + k] * basis[(k * Cc + c) * INc + i];
    W1[t] = s;
}

// ---------------- W2[g,f] = sum_k qk_coeff[rh,k]*qk_basis[k,o,f] ----------------
__global__ void build_w2(const float* __restrict__ coeff, const float* __restrict__ basis,
                         float* __restrict__ W2) {
    long long t = (long long)blockIdx.x * blockDim.x + threadIdx.x;  // [0, Gc*Fc)
    if (t >= (long long)Gc * Fc) return;
    int g = (int)(t / Fc), f = (int)(t % Fc);
    int rh = g >> 6, o = g & 63;  // 2A = 64
    float s = 0.f;
#pragma unroll
    for (int k = 0; k < NBc; ++k)
        s += coeff[rh * NBc + k] * basis[((k * (2 * Ac)) + o) * Fc + f];
    W2[(size_t)g * Fc + f] = s;
}

// ---------------- register-blocked fp32 WMMA GEMM -----------------------------------
// C[M,N] = A[M,K] * Bt[N,K]^T + bias[N].
// One wave computes a 64x32 tile as a 4x2 grid of 16x16 WMMA tiles: 8 accumulators
// (64 VGPRs) + 6 fragment regs stays under the compiler's VGPR budget (the 4x4
// variant spilled accumulators to scratch). Per K-step: 4 A-frag + 2 B-frag b64 loads
// feed 8 v_wmma_f32_16x16x4_f32. K,N compile-time so fragment offsets fold into the
// 24-bit load immediates and loads clause.
template <int K, int N>
__global__ void __launch_bounds__(256, 1)
gemm_wmma_f32_blk(const float* __restrict__ A, const float* __restrict__ Bt,
                  const float* __restrict__ bias, float* __restrict__ C, int M) {
    int wave = (blockIdx.x * blockDim.x + threadIdx.x) >> 5;
    int lane = threadIdx.x & 31;
    constexpr int NT = N / 32;
    int tm = (wave / NT) * 64;
    int tn = (wave % NT) * 32;
    if (tm >= M) return;

    int half = lane >> 4;   // selects K pair {0,1} vs {2,3} within each k-step
    int l16  = lane & 15;

    const float* arow = A  + (size_t)(tm + l16) * K + 2 * half;
    const float* brow = Bt + (size_t)(tn + l16) * K + 2 * half;

    v8f acc[4][2] = {};
#pragma unroll 2
    for (int k0 = 0; k0 < K; k0 += 4) {
        v2f a[4], b[2];
#pragma unroll
        for (int i = 0; i < 4; ++i)
            a[i] = *(const v2f*)(arow + (size_t)i * 16 * K + k0);
#pragma unroll
        for (int j = 0; j < 2; ++j)
            b[j] = *(const v2f*)(brow + (size_t)j * 16 * K + k0);
#pragma unroll
        for (int i = 0; i < 4; ++i)
#pragma unroll
            for (int j = 0; j < 2; ++j)
                acc[i][j] = __builtin_amdgcn_wmma_f32_16x16x4_f32(
                    /*neg_a=*/false, a[i], /*neg_b=*/false, b[j],
                    /*c_mod=*/(short)0, acc[i][j], /*reuse_a=*/false, /*reuse_b=*/false);
    }

    float bv[2];
#pragma unroll
    for (int j = 0; j < 2; ++j) bv[j] = bias[tn + j * 16 + l16];

#pragma unroll
    for (int i = 0; i < 4; ++i)
#pragma unroll
        for (int r = 0; r < 8; ++r) {
            int row = tm + i * 16 + r + 8 * half;   // C/D layout: vgpr r -> M=r / M=r+8
            float* crow = C + (size_t)row * N + tn + l16;
#pragma unroll
            for (int j = 0; j < 2; ++j)
                crow[j * 16] = acc[i][j][r] + bv[j];
        }
}

// ---------------- pass 1: per-(edge,h) logits + segment max ----------------
__global__ void edge_logits_max(const float* __restrict__ QK,
                                const int* __restrict__ eb, const int* __restrict__ ed,
                                const int* __restrict__ es, const int* __restrict__ er,
                                float* __restrict__ logits, unsigned* __restrict__ mmax,
                                int E) {
    int t = blockIdx.x * blockDim.x + threadIdx.x;
    int e = t >> 2, h = t & (Hc - 1);
    if (e >= E) return;
    int b = eb[e], dst = ed[e], src = es[e], rel = er[e];
    size_t base = (size_t)(rel * Hc + h) * (2 * Ac);
    const float4* q4 = (const float4*)(QK + (size_t)(b * Nc + dst) * Gc + base);
    const float4* k4 = (const float4*)(QK + (size_t)(b * Nc + src) * Gc + base + Ac);
    float s = 0.f;
#pragma unroll
    for (int i = 0; i < Ac / 4; ++i) {
        float4 q = q4[i], k = k4[i];
        s += q.x * k.x + q.y * k.y + q.z * k.z + q.w * k.w;
    }
    logits[t] = s;
    int seg = (b * Nc + dst) * Rc + rel;
    atomicMax(&mmax[seg * Hc + h], flipf(s));
}

// ---------------- pass 2: exp(logit - max), segment denominator ----------------
__global__ void edge_exp_sum(const int* __restrict__ eb, const int* __restrict__ ed,
                             const int* __restrict__ er,
                             float* __restrict__ logits, const unsigned* __restrict__ mmax,
                             float* __restrict__ den, int E) {
    int t = blockIdx.x * blockDim.x + threadIdx.x;
    int e = t >> 2, h = t & (Hc - 1);
    if (e >= E) return;
    int seg = (eb[e] * Nc + ed[e]) * Rc + er[e];
    float m  = unflipf(mmax[seg * Hc + h]);
    float ex = expf(logits[t] - m);
    logits[t] = ex;
    atomicAdd(&den[seg * Hc + h], ex);
}

// ---------------- pass 3: wave per edge, lane = channel c, atomic scatter to out ---------
__global__ void edge_scatter(const float* __restrict__ V,
                             const int* __restrict__ eb, const int* __restrict__ ed,
                             const int* __restrict__ es, const int* __restrict__ er,
                             const float* __restrict__ exv, const float* __restrict__ den,
                             float* __restrict__ out, int E) {
    int wid  = (blockIdx.x * blockDim.x + threadIdx.x) >> 5;
    int lane = threadIdx.x & 31;                  // lane == channel c (C=32)
    if (wid >= E) return;
    int b = eb[wid], dst = ed[wid], src = es[wid], rel = er[wid];
    int seg = (b * Nc + dst) * Rc + rel;
    const float* vrow = V + (size_t)(b * Nc + src) * Fc;
    float* orow = out + (size_t)(b * Nc + dst) * (Hc * Cc);
#pragma unroll
    for (int h = 0; h < Hc; ++h) {
        float attn = exv[wid * Hc + h] / den[seg * Hc + h];
        float val  = attn * vrow[(size_t)(rel * Hc + h) * Cc + lane];
        atomicAdd(orow + h * Cc + lane, val);
    }
}

// ---------------- launch ----------------
extern "C" void kernel_launch(void* const* d_in, const int* in_sizes, int n_in,
                              void* d_out, int out_size, void* d_ws, size_t ws_size,
                              hipStream_t stream) {
    const float* node       = (const float*)d_in[0];
    const int*   eb         = (const int*)d_in[1];
    const int*   ed         = (const int*)d_in[2];
    const int*   es         = (const int*)d_in[3];
    const int*   er         = (const int*)d_in[4];
    const float* proj_basis = (const float*)d_in[5];
    const float* proj_coeff = (const float*)d_in[6];
    const float* proj_bias  = (const float*)d_in[7];
    const float* qk_basis   = (const float*)d_in[8];
    const float* qk_coeff   = (const float*)d_in[9];
    const float* qk_bias    = (const float*)d_in[10];
    int E = in_sizes[1];
    float* out = (float*)d_out;

    // workspace carve-out
    char* p = (char*)d_ws;
    float*    W1     = (float*)p;    p += sizeof(float)    * (size_t)Fc * INc;      // 0.5 MB
    float*    W2     = (float*)p;    p += sizeof(float)    * (size_t)Gc * Fc;       // 8 MB
    float*    V      = (float*)p;    p += sizeof(float)    * (size_t)Mrows * Fc;    // 160 MB
    float*    QK     = (float*)p;    p += sizeof(float)    * (size_t)Mrows * Gc;    // 320 MB
    float*    logits = (float*)p;    p += sizeof(float)    * (size_t)E * Hc;        // 16 MB
    unsigned* mmax   = (unsigned*)p; p += sizeof(unsigned) * (size_t)Sc * Hc;       // 5 MB
    float*    den    = (float*)p;                                                   // 5 MB

    dim3 blk(256);

    // init out / den / mmax (out is largest: Mrows*H*C = 5.12M elems)
    init_bufs<<<(Mrows * Hc * Cc + 255) / 256, blk, 0, stream>>>(
        out, den, mmax, Mrows * Hc * Cc, Sc * Hc);

    // tiny basis contractions
    build_w1<<<(Fc * INc + 255) / 256, blk, 0, stream>>>(proj_coeff, proj_basis, W1);
    build_w2<<<(int)(((size_t)Gc * Fc + 255) / 256), blk, 0, stream>>>(qk_coeff, qk_basis, W2);

    // V = node_states(40000x128) * W1^T + proj_bias  -> (40000x1024)
    {
        int waves = (Mrows / 64) * (Fc / 32);      // 625 * 32
        gemm_wmma_f32_blk<INc, Fc><<<(waves * 32 + 255) / 256, blk, 0, stream>>>(
            node, W1, proj_bias, V, Mrows);
    }
    // QK = V(40000x1024) * W2^T + qk_bias  -> (40000x2048)
    {
        int waves = (Mrows / 64) * (Gc / 32);      // 625 * 64
        gemm_wmma_f32_blk<Fc, Gc><<<(waves * 32 + 255) / 256, blk, 0, stream>>>(
            V, W2, qk_bias, QK, Mrows);
    }

    // edge phase
    int tEH = E * Hc;
    edge_logits_max<<<(tEH + 255) / 256, blk, 0, stream>>>(QK, eb, ed, es, er,
                                                           logits, mmax, E);
    edge_exp_sum<<<(tEH + 255) / 256, blk, 0, stream>>>(eb, ed, er, logits, mmax, den, E);
    edge_scatter<<<(int)(((size_t)E * 32 + 255) / 256), blk, 0, stream>>>(
        V, eb, ed, es, er, logits, den, out, E);
}